// Prototypes_21732534518436
// MI455X (gfx1250) — compile-verified
//
#include <hip/hip_runtime.h>
#include <stdint.h>

// Problem constants (fixed by the reference).
#define C_CLS      1000
#define D_DIM      128
#define DH         64          // half of D handled per block (blockIdx.y selects half)
#define ACC_STRIDE 65          // skewed stride -> LDS bank index depends on class id
#define TILE_R     64          // feature rows staged per tile
#define CHUNKS     256         // row chunks (gridDim.x)
#define THREADS    512         // 16 wave32 waves -> 4 waves/SIMD32 (1 WG/WGP at 297KB LDS)
#define F4_PER_TILE   ((TILE_R * DH) / 4)          // 1024 float4 per tile
#define F4_PER_THREAD (F4_PER_TILE / THREADS)      // 2 async b128 issues per lane/tile
#define ROWS_PER_PASS (THREADS / 16)               // 32 rows accumulated per pass

#ifndef __has_builtin
#define __has_builtin(x) 0
#endif

#if __has_builtin(__builtin_amdgcn_global_load_async_to_lds_b128) && \
    __has_builtin(__builtin_amdgcn_s_wait_asynccnt)
#define USE_ASYNC 1
#else
#define USE_ASYNC 0
#endif

// Pointer types the async builtin wants: v4i32 in explicit address spaces.
typedef int v4i __attribute__((ext_vector_type(4)));
typedef __attribute__((address_space(1))) v4i g1_v4i;   // global
typedef __attribute__((address_space(3))) v4i l3_v4i;   // LDS

// Stage one TILE_R x DH tile of feats (row stride D_DIM) into an LDS buffer.
// Each lane moves F4_PER_THREAD float4s via the CDNA5 async-copy path
// (tracked by ASYNCcnt); fallback is a plain b128 load + ds store.
__device__ __forceinline__ void stage_tile(const float* __restrict__ gbase,
                                           float* lbuf, int tid) {
#pragma unroll
  for (int k = 0; k < F4_PER_THREAD; ++k) {
    const int idx4 = tid + k * THREADS;   // 0..1023
    const int row  = idx4 >> 4;           // 16 float4 per row
    const int c4   = idx4 & 15;
    const float* g = gbase + row * D_DIM + c4 * 4;
    float* l = lbuf + idx4 * 4;
#if USE_ASYNC
    // generic->AS casts via integer reinterpret (flat LDS ptr low 32 bits are
    // the LDS byte offset; flat global ptr value == global address).
    g1_v4i* gp = (g1_v4i*)(uintptr_t)g;
    l3_v4i* lp = (l3_v4i*)(uint32_t)(uintptr_t)l;
    __builtin_amdgcn_global_load_async_to_lds_b128(gp, lp, 0, 0);
#else
    *(float4*)l = *(const float4*)g;
#endif
  }
}

__global__ __launch_bounds__(THREADS) void proto_accum(
    const float* __restrict__ feats, const int* __restrict__ labels,
    const float* __restrict__ weights, float* __restrict__ gAcc,
    float* __restrict__ gWsum, int rowsPerChunk) {
  // ~297 KB static LDS: legal on CDNA5's 320 KB WGP LDS, impossible on CDNA4.
  __shared__ float accLds[C_CLS * ACC_STRIDE];          // 260 000 B
  __shared__ float wsumLds[C_CLS];                      //   4 000 B
  __shared__ __align__(16) float stage[2][TILE_R * DH]; //  32 768 B

  const int tid      = threadIdx.x;
  const int half     = blockIdx.y;                 // which 64 columns of D
  const int rowBase0 = blockIdx.x * rowsPerChunk;
  const int ntiles   = rowsPerChunk / TILE_R;

  for (int i = tid; i < C_CLS * ACC_STRIDE; i += THREADS) accLds[i] = 0.0f;
  for (int i = tid; i < C_CLS; i += THREADS) wsumLds[i] = 0.0f;
  __syncthreads();

  const float* gbase0 = feats + (size_t)rowBase0 * D_DIM + half * DH;

  stage_tile(gbase0, stage[0], tid);  // prime the pipeline

  for (int t = 0; t < ntiles; ++t) {
    const int buf = t & 1;
    const bool hasNext = (t + 1 < ntiles);
    if (hasNext)
      stage_tile(gbase0 + (size_t)(t + 1) * TILE_R * D_DIM, stage[buf ^ 1], tid);
#if USE_ASYNC
    // Async loads complete in order: cnt<=F4_PER_THREAD means the current
    // tile's copies are done while the next tile's stay in flight.
    if (hasNext) __builtin_amdgcn_s_wait_asynccnt(F4_PER_THREAD);
    else         __builtin_amdgcn_s_wait_asynccnt(0);
#endif
    __syncthreads();  // all waves' copies of stage[buf] visible

    const int rowBase = rowBase0 + t * TILE_R;
    const int r_sub = tid >> 4;   // 16 lanes per feature row
    const int c4    = tid & 15;   // float4 column within the half-row
#pragma unroll
    for (int rr = 0; rr < TILE_R; rr += ROWS_PER_PASS) {
      const int row = rr + r_sub;
      const int gr  = rowBase + row;
      const int lab = labels[gr];
      const float w = weights[gr];
      const float4 v = *(const float4*)(&stage[buf][row * DH + c4 * 4]);
      float* dst = &accLds[lab * ACC_STRIDE + c4 * 4];
      atomicAdd(dst + 0, w * v.x);   // ds_add_f32, conflict-free via skew
      atomicAdd(dst + 1, w * v.y);
      atomicAdd(dst + 2, w * v.z);
      atomicAdd(dst + 3, w * v.w);
      if (half == 0 && c4 == 0) atomicAdd(&wsumLds[lab], w);
    }
    __syncthreads();  // protect stage[buf] before it is re-issued
  }

  // Flush per-block partials: 64K global f32 atomics per block instead of
  // 64 cols * rowsPerChunk.
  for (int i = tid; i < C_CLS * DH; i += THREADS) {
    const int c = i >> 6, col = i & 63;
    atomicAdd(&gAcc[c * D_DIM + half * DH + col], accLds[c * ACC_STRIDE + col]);
  }
  if (half == 0)
    for (int c = tid; c < C_CLS; c += THREADS) atomicAdd(&gWsum[c], wsumLds[c]);
}

__global__ void proto_finalize(const float* __restrict__ proto,
                               const float* __restrict__ gAcc,
                               const float* __restrict__ gWsum,
                               float* __restrict__ out) {
  const int i = blockIdx.x * blockDim.x + threadIdx.x;
  if (i >= C_CLS * D_DIM) return;
  const int c = i >> 7;  // D = 128
  const float ws  = gWsum[c];
  const float p   = proto[i];
  const float vec = gAcc[i] / fmaxf(ws, 1e-8f);
  out[i] = (ws > 0.0f) ? (0.95f * p + 0.05f * vec) : p;
}

extern "C" void kernel_launch(void* const* d_in, const int* in_sizes, int n_in,
                              void* d_out, int out_size, void* d_ws, size_t ws_size,
                              hipStream_t stream) {
  (void)n_in; (void)out_size; (void)ws_size;
  const float* feats   = (const float*)d_in[0];
  const int*   labels  = (const int*)d_in[1];
  const float* weights = (const float*)d_in[2];
  const float* proto   = (const float*)d_in[3];
  float* out   = (float*)d_out;
  float* gAcc  = (float*)d_ws;                // C*D weighted feature sums
  float* gWsum = gAcc + C_CLS * D_DIM;        // C weight sums

  const int n = in_sizes[1];                  // N (labels count)
  const int rowsPerChunk = n / CHUNKS;        // 4096 for N=1M; TILE_R divides it

  (void)hipMemsetAsync(d_ws, 0,
                       (size_t)(C_CLS * D_DIM + C_CLS) * sizeof(float), stream);

  dim3 grid(CHUNKS, 2), block(THREADS);
  proto_accum<<<grid, block, 0, stream>>>(feats, labels, weights, gAcc, gWsum,
                                          rowsPerChunk);

  const int total = C_CLS * D_DIM;
  proto_finalize<<<(total + 255) / 256, 256, 0, stream>>>(proto, gAcc, gWsum, out);
}